// GNNModel_33655363732257
// MI455X (gfx1250) — compile-verified
//
#include <hip/hip_runtime.h>

// GCN 2-layer forward, MI455X (gfx1250, wave32).
// Memory/atomics-bound: edge scatter moves ~2.4GB -> ~100us @ 23.3 TB/s HBM.
// Dense GEMMs use V_WMMA_F32_16X16X4_F32 (exact f32 matrix path), full tiles only;
// remainder rows (N%16) go through a scalar tail kernel launched only if needed.

typedef __attribute__((ext_vector_type(2))) float v2f;
typedef __attribute__((ext_vector_type(8))) float v8f;

__device__ __forceinline__ void gadd(float* p, float v) {
  // hardware global_atomic_add_f32 (no CAS loop)
  unsafeAtomicAdd(p, v);
}

// ---------------- utility fills ----------------
__global__ void k_fill(float* __restrict__ p, long n, float v) {
  long i = (long)blockIdx.x * blockDim.x + threadIdx.x;
  if (i < n) p[i] = v;
}

// ---------------- degree + dinv ----------------
__global__ void k_deg(const int* __restrict__ dst, int E, float* __restrict__ deg) {
  int e = blockIdx.x * blockDim.x + threadIdx.x;
  if (e < E) gadd(&deg[dst[e]], 1.0f);
}

__global__ void k_rsqrt(float* __restrict__ deg, int n) {
  int i = blockIdx.x * blockDim.x + threadIdx.x;
  if (i < n) deg[i] = rsqrtf(deg[i]);   // deg >= 1 (self loop)
}

// ---------------- GEMM1: x[N,4] @ W1[4,64] -> hW1[N,64], WMMA f32 16x16x4 ----------------
// Full 16-row tiles only; no bounds checks inside.
__global__ void k_gemm1(const float* __restrict__ x, const float* __restrict__ W1,
                        float* __restrict__ hW1, int n_tiles) {
  int wave = blockIdx.x * (blockDim.x >> 5) + (threadIdx.x >> 5);
  if (wave >= n_tiles) return;                   // wave-uniform: EXEC stays full
  int lane = threadIdx.x & 31;
  int m = lane & 15;                             // M (A) / N (B,D) index within tile
  int g = lane >> 4;                             // lane group
  int row0 = wave * 16;

  const float* xr = x + (row0 + m) * 4 + 2 * g;
  v2f a; a.x = xr[0]; a.y = xr[1];               // A[m, 2g], A[m, 2g+1]

  const float* wb = W1 + 2 * g * 64 + m;         // B base for this lane

#pragma unroll
  for (int nt = 0; nt < 4; ++nt) {
    int n0 = nt * 16;
    v2f b;
    b.x = wb[n0];                                // B[2g,   n0+m]
    b.y = wb[n0 + 64];                           // B[2g+1, n0+m]
    v8f c = {};
    c = __builtin_amdgcn_wmma_f32_16x16x4_f32(false, a, false, b, (short)0, c, false, false);
    // D: vgpr r -> row row0 + r + 8g, col n0 + m  (clause of immediate-offset stores)
    float* d0 = hW1 + (row0 + 8 * g) * 64 + n0 + m;
#pragma unroll
    for (int r = 0; r < 8; ++r) d0[r * 64] = c[r];
  }
}

// ---------------- GEMM2: h1[N,64] @ W2[64,32] -> hW2[N,32], 16 k-steps x 2 n-tiles ----------------
__global__ void k_gemm2(const float* __restrict__ h1, const float* __restrict__ W2,
                        float* __restrict__ hW2, int n_tiles) {
  int wave = blockIdx.x * (blockDim.x >> 5) + (threadIdx.x >> 5);
  if (wave >= n_tiles) return;                   // wave-uniform
  int lane = threadIdx.x & 31;
  int m = lane & 15;
  int g = lane >> 4;
  int row0 = wave * 16;

  const float* arow = h1 + (row0 + m) * 64 + 2 * g;  // A pairs at arow[4*kk], arow[4*kk+1]
  const float* wb = W2 + 2 * g * 32 + m;             // B pairs at wb[kk*128 + {0,32}] (+16 tile 1)

  v8f c0 = {}, c1 = {};
#pragma unroll
  for (int kk = 0; kk < 16; ++kk) {
    v2f a; a.x = arow[4 * kk]; a.y = arow[4 * kk + 1];
    v2f b0, b1v;
    b0.x  = wb[kk * 128];       b0.y  = wb[kk * 128 + 32];
    b1v.x = wb[kk * 128 + 16];  b1v.y = wb[kk * 128 + 48];
    c0 = __builtin_amdgcn_wmma_f32_16x16x4_f32(false, a, false, b0,  (short)0, c0, false, false);
    c1 = __builtin_amdgcn_wmma_f32_16x16x4_f32(false, a, false, b1v, (short)0, c1, false, false);
  }

  float* d0 = hW2 + (row0 + 8 * g) * 32 + m;     // D epilogue, immediate offsets
#pragma unroll
  for (int r = 0; r < 8; ++r) {
    d0[r * 32]      = c0[r];
    d0[r * 32 + 16] = c1[r];
  }
}

// ---------------- scalar GEMM tail for rows in [row0, row0+nrows) ----------------
// Launched only when N % 16 != 0 (never for N=100000).
__global__ void k_gemm_tail(const float* __restrict__ A, const float* __restrict__ W,
                            float* __restrict__ out, int row0, int nrows, int K, int M) {
  int tid = blockIdx.x * blockDim.x + threadIdx.x;
  if (tid >= nrows * M) return;
  int row = row0 + tid / M;
  int col = tid - (tid / M) * M;
  float acc = 0.f;
  for (int k = 0; k < K; ++k) acc = fmaf(A[row * K + k], W[k * M + col], acc);
  out[row * M + col] = acc;
}

// ---------------- scatter layer1: agg1[dst] += hW1[src]*norm (float4 atomics) ----------------
__global__ void k_scatter1(const int* __restrict__ src, const int* __restrict__ dst,
                           const float* __restrict__ dinv, const float* __restrict__ hW1,
                           float* __restrict__ agg1, int E) {
  unsigned tid = blockIdx.x * blockDim.x + threadIdx.x;   // < E*16 = 51.2M
  unsigned e = tid >> 4;
  if (e >= (unsigned)E) return;
  int f0 = (int)(tid & 15u) * 4;
  int s = src[e], d = dst[e];
  float norm = dinv[s] * dinv[d];
  const float4 v = *(const float4*)(hW1 + s * 64 + f0);
  float* p = agg1 + d * 64 + f0;
  gadd(p + 0, v.x * norm);
  gadd(p + 1, v.y * norm);
  gadd(p + 2, v.z * norm);
  gadd(p + 3, v.w * norm);
}

// ---------------- self-loop + bias + relu: h1 = relu(agg1 + hW1*dinv^2 + b1) ----------------
__global__ void k_relu_bias(float* __restrict__ agg1, const float* __restrict__ hW1,
                            const float* __restrict__ dinv, const float* __restrict__ b1, int N) {
  unsigned tid = blockIdx.x * blockDim.x + threadIdx.x;   // < N*16
  int i = (int)(tid >> 4);
  if (i >= N) return;
  int f0 = (int)(tid & 15u) * 4;
  float di = dinv[i], n2 = di * di;
  float4 a = *(float4*)(agg1 + i * 64 + f0);
  float4 h = *(const float4*)(hW1 + i * 64 + f0);
  float4 bb = *(const float4*)(b1 + f0);
  a.x = fmaxf(fmaf(h.x, n2, a.x) + bb.x, 0.f);
  a.y = fmaxf(fmaf(h.y, n2, a.y) + bb.y, 0.f);
  a.z = fmaxf(fmaf(h.z, n2, a.z) + bb.z, 0.f);
  a.w = fmaxf(fmaf(h.w, n2, a.w) + bb.w, 0.f);
  *(float4*)(agg1 + i * 64 + f0) = a;
}

// ---------------- scatter layer2: agg2[dst] += hW2[src]*norm ----------------
__global__ void k_scatter2(const int* __restrict__ src, const int* __restrict__ dst,
                           const float* __restrict__ dinv, const float* __restrict__ hW2,
                           float* __restrict__ agg2, int E) {
  unsigned tid = blockIdx.x * blockDim.x + threadIdx.x;   // < E*8
  unsigned e = tid >> 3;
  if (e >= (unsigned)E) return;
  int f0 = (int)(tid & 7u) * 4;
  int s = src[e], d = dst[e];
  float norm = dinv[s] * dinv[d];
  const float4 v = *(const float4*)(hW2 + s * 32 + f0);
  float* p = agg2 + d * 32 + f0;
  gadd(p + 0, v.x * norm);
  gadd(p + 1, v.y * norm);
  gadd(p + 2, v.z * norm);
  gadd(p + 3, v.w * norm);
}

// ---------------- reduction: sum32[f] = sum_i (agg2[i,f] + hW2[i,f]*dinv[i]^2) ----------------
__global__ void k_reduce(const float* __restrict__ agg2, const float* __restrict__ hW2,
                         const float* __restrict__ dinv, float* __restrict__ sum32, int N) {
  __shared__ float s[32];
  if (threadIdx.x < 32) s[threadIdx.x] = 0.f;
  __syncthreads();
  int f = threadIdx.x & 31;                      // stride is a multiple of 32 -> f constant
  int total = N * 32;
  int stride = gridDim.x * blockDim.x;
  float acc = 0.f;
  for (int idx = blockIdx.x * blockDim.x + threadIdx.x; idx < total; idx += stride) {
    int i = idx >> 5;
    float di = dinv[i];
    acc = fmaf(hW2[idx], di * di, acc) + agg2[idx];
  }
  atomicAdd(&s[f], acc);                         // ds_add_f32
  __syncthreads();
  if (threadIdx.x < 32) gadd(&sum32[threadIdx.x], s[threadIdx.x]);
}

__global__ void k_finalize(const float* __restrict__ sum32, const float* __restrict__ b2,
                           float* __restrict__ out, float invN) {
  int f = threadIdx.x;
  if (f < 32) out[f] = sum32[f] * invN + b2[f];
}

// ---------------- host-side launch ----------------
static inline unsigned cdiv(long a, long b) { return (unsigned)((a + b - 1) / b); }

extern "C" void kernel_launch(void* const* d_in, const int* in_sizes, int n_in,
                              void* d_out, int out_size, void* d_ws, size_t ws_size,
                              hipStream_t stream) {
  const float* x  = (const float*)d_in[0];
  const int*   ei = (const int*)d_in[1];   // edge_index [2, E] (int32 per harness convention)
  const float* W1 = (const float*)d_in[2];
  const float* b1 = (const float*)d_in[3];
  const float* W2 = (const float*)d_in[4];
  const float* b2 = (const float*)d_in[5];
  float* out = (float*)d_out;

  const long N = in_sizes[0] / 4;
  const long E = in_sizes[1] / 2;
  const int* src = ei;
  const int* dst = ei + E;

  float* ws    = (float*)d_ws;
  float* deg   = ws;                    // N floats (becomes dinv in place)
  float* bufA  = ws + N;                // 64N (hW1, reused as hW2)
  float* bufB  = ws + N + 64 * N;       // 64N (agg1 -> h1)
  float* agg2  = ws + N + 128 * N;      // 32N
  float* sum32 = ws + N + 160 * N;      // 32

  const int T = 256;
  const int tiles = (int)(N / 16);      // full WMMA tiles
  const int rem   = (int)(N % 16);      // scalar tail rows (0 for N=100000)
  const int rowT  = (int)(N - rem);

  // init (re-zero every call: harness does not re-poison between replays)
  k_fill<<<cdiv(N, T), T, 0, stream>>>(deg, N, 1.0f);            // self loop
  k_fill<<<cdiv(64 * N, T), T, 0, stream>>>(bufB, 64 * N, 0.0f);
  k_fill<<<cdiv(32 * N, T), T, 0, stream>>>(agg2, 32 * N, 0.0f);
  k_fill<<<1, 32, 0, stream>>>(sum32, 32, 0.0f);

  // degree + dinv
  k_deg<<<cdiv(E, T), T, 0, stream>>>(dst, (int)E, deg);
  k_rsqrt<<<cdiv(N, T), T, 0, stream>>>(deg, (int)N);

  // layer 1
  if (tiles) k_gemm1<<<cdiv(tiles, T / 32), T, 0, stream>>>(x, W1, bufA, tiles);
  if (rem)   k_gemm_tail<<<cdiv(rem * 64, 64), 64, 0, stream>>>(x, W1, bufA, rowT, rem, 4, 64);
  k_scatter1<<<cdiv(E * 16, T), T, 0, stream>>>(src, dst, deg, bufA, bufB, (int)E);
  k_relu_bias<<<cdiv(N * 16, T), T, 0, stream>>>(bufB, bufA, deg, b1, (int)N);

  // layer 2 (bufA reused for hW2)
  if (tiles) k_gemm2<<<cdiv(tiles, T / 32), T, 0, stream>>>(bufB, W2, bufA, tiles);
  if (rem)   k_gemm_tail<<<cdiv(rem * 32, 64), 64, 0, stream>>>(bufB, W2, bufA, rowT, rem, 64, 32);
  k_scatter2<<<cdiv(E * 8, T), T, 0, stream>>>(src, dst, deg, bufA, agg2, (int)E);

  // mean over nodes (+ self-loop term) then bias
  k_reduce<<<1536, T, 0, stream>>>(agg2, bufA, deg, sum32, (int)N);
  k_finalize<<<1, 32, 0, stream>>>(sum32, b2, out, 1.0f / (float)N);
}